// GaussianAttention_46832323395600
// MI455X (gfx1250) — compile-verified
//
#include <hip/hip_runtime.h>

typedef __attribute__((ext_vector_type(2))) float v2f;
typedef __attribute__((ext_vector_type(8))) float v8f;

#define Bsz 256
#define Tsz 1024
#define Csz 512
#define Hsz 1024
#define Ksz 10

// ---------------------------------------------------------------------------
// Kernel 1: params = exp(h_t @ W.T + bias); split into alpha/beta/kappa.
// One wave (32 lanes) computes a 16(batch) x 16(j) tile with
// V_WMMA_F32_16X16X4_F32, reducing over H=1024 in K-steps of 4.
// grid = 32 blocks (16 batch-tiles x 2 j-tiles), block = 32 (one wave).
// ---------------------------------------------------------------------------
__global__ __launch_bounds__(32) void ga_params_wmma(
    const float* __restrict__ h,      // [B,H]
    const float* __restrict__ W,      // [3K,H]
    const float* __restrict__ bias,   // [3K]
    const float* __restrict__ k_tm1,  // [B,K]
    float* __restrict__ alpha,        // ws [B,K]
    float* __restrict__ beta,         // ws [B,K]
    float* __restrict__ kappa,        // ws [B,K]
    float* __restrict__ kappa_out)    // d_out tail [B,K]
{
    const int lane = threadIdx.x;
    const int m    = lane & 15;       // A row / D col index base
    const int half = lane >> 4;       // 0: K/N rows 0..., 1: upper half
    const int b0   = (blockIdx.x >> 1) << 4;   // batch tile origin
    const int j0   = (blockIdx.x & 1) << 4;    // j tile origin (0 or 16)

    int jb = j0 + m;                  // B-matrix column -> W row
    if (jb > 3 * Ksz - 1) jb = 3 * Ksz - 1;    // clamp (masked at writeback)

    const float* __restrict__ arow = h + (size_t)(b0 + m) * Hsz;
    const float* __restrict__ brow = W + (size_t)jb * Hsz;

    v8f acc = {};
#pragma unroll 8
    for (int k0 = 0; k0 < Hsz; k0 += 4) {
        // A[16x4]: lane m=row, VGPR pair holds K = 2*half + {0,1}
        v2f afrag = ((const v2f*)(arow + k0))[half];
        // B[4x16]: lane n=col, VGPR pair holds K = 2*half + {0,1}
        v2f bfrag = ((const v2f*)(brow + k0))[half];
        acc = __builtin_amdgcn_wmma_f32_16x16x4_f32(
            /*neg_a=*/false, afrag, /*neg_b=*/false, bfrag,
            /*c_mod=*/(short)0, acc, /*reuse_a=*/false, /*reuse_b=*/false);
    }

    const int j = j0 + m;             // output column for this lane
#pragma unroll
    for (int v = 0; v < 8; ++v) {
        const int row = b0 + v + half * 8;     // batch index
        if (j < 3 * Ksz) {
            float val = __expf(acc[v] + bias[j]);
            if (j < Ksz) {
                alpha[row * Ksz + j] = val;
            } else if (j < 2 * Ksz) {
                beta[row * Ksz + (j - Ksz)] = val;
            } else {
                const int kk = j - 2 * Ksz;
                float kp = val + k_tm1[row * Ksz + kk];
                kappa[row * Ksz + kk]     = kp;
                kappa_out[row * Ksz + kk] = kp;
            }
        }
    }
}

// ---------------------------------------------------------------------------
// Kernel 2: phi[b,t] = sum_k alpha*exp(-beta*(kappa-t)^2).
// grid = (B, T/256), block = 256. Per-block params are uniform -> scalar
// loads; exp on VALU trans pipe. Negligible cost.
// ---------------------------------------------------------------------------
__global__ __launch_bounds__(256) void ga_phi(
    const float* __restrict__ alpha, const float* __restrict__ beta,
    const float* __restrict__ kappa, float* __restrict__ phi_out)
{
    const int b = blockIdx.x;
    const int t = blockIdx.y * 256 + threadIdx.x;
    const float u = (float)t;
    const float* __restrict__ a  = alpha + b * Ksz;
    const float* __restrict__ be = beta  + b * Ksz;
    const float* __restrict__ kp = kappa + b * Ksz;
    float s = 0.0f;
#pragma unroll
    for (int k = 0; k < Ksz; ++k) {
        float d = kp[k] - u;
        s = __builtin_fmaf(a[k], __expf(-be[k] * d * d), s);
    }
    phi_out[(size_t)b * Tsz + t] = s;
}

// ---------------------------------------------------------------------------
// Zero-init for w (accumulated with atomics).
// ---------------------------------------------------------------------------
__global__ __launch_bounds__(256) void ga_zero_w(float* __restrict__ w)
{
    const int i = blockIdx.x * 256 + threadIdx.x;
    if (i < Bsz * Csz) w[i] = 0.0f;
}

// ---------------------------------------------------------------------------
// Kernel 3 (bandwidth-dominant): w[b,c] = sum_t phi[b,t]*ctx[b,t,c].
// Streams ctx (512 MB) exactly once: 128 threads x float4 cover the full
// C=512 row per load-step (global_load_b128, fully coalesced). T split into
// 4 segments across blockIdx.y for occupancy (1024 blocks); partial sums
// combined with global_atomic_add_f32 (2 MB of atomics vs 512 MB streamed).
// phi staged in LDS once per block; prefetch runs ahead of the stream.
// ---------------------------------------------------------------------------
__global__ __launch_bounds__(128) void ga_wsum(
    const float* __restrict__ phi,    // [B,T] (in d_out)
    const float* __restrict__ ctx,    // [B,T,C]
    float* __restrict__ w)            // [B,C]
{
    __shared__ float sphi[256];
    const int b   = blockIdx.x;
    const int t0  = blockIdx.y * 256;
    const int tid = threadIdx.x;

    sphi[tid]       = phi[(size_t)b * Tsz + t0 + tid];
    sphi[tid + 128] = phi[(size_t)b * Tsz + t0 + tid + 128];
    __syncthreads();

    // float4 lane base: row (b,t0), column tid*4
    const float4* __restrict__ base =
        (const float4*)(ctx + ((size_t)b * Tsz + t0) * Csz) + tid;

    float ax = 0.0f, ay = 0.0f, az = 0.0f, aw = 0.0f;
#pragma unroll 4
    for (int tt = 0; tt < 256; ++tt) {
        float4 v = base[tt * (Csz / 4)];
        __builtin_prefetch((const void*)(base + (tt + 8) * (Csz / 4)), 0, 0);
        const float p = sphi[tt];
        ax = __builtin_fmaf(p, v.x, ax);
        ay = __builtin_fmaf(p, v.y, ay);
        az = __builtin_fmaf(p, v.z, az);
        aw = __builtin_fmaf(p, v.w, aw);
    }

    float* wrow = w + (size_t)b * Csz + tid * 4;
    atomicAdd(wrow + 0, ax);
    atomicAdd(wrow + 1, ay);
    atomicAdd(wrow + 2, az);
    atomicAdd(wrow + 3, aw);
}

// ---------------------------------------------------------------------------
extern "C" void kernel_launch(void* const* d_in, const int* in_sizes, int n_in,
                              void* d_out, int out_size, void* d_ws, size_t ws_size,
                              hipStream_t stream)
{
    const float* h_t   = (const float*)d_in[0];   // [B,H]
    const float* k_tm1 = (const float*)d_in[1];   // [B,1,K]
    const float* ctx   = (const float*)d_in[2];   // [B,T,C]
    const float* W     = (const float*)d_in[3];   // [3K,H]
    const float* bias  = (const float*)d_in[4];   // [3K]

    float* out       = (float*)d_out;
    float* w_out     = out;                               // B*C   = 131072
    float* phi_out   = out + (size_t)Bsz * Csz;           // B*T   = 262144
    float* kappa_out = out + (size_t)Bsz * Csz + (size_t)Bsz * Tsz; // B*K

    float* ws    = (float*)d_ws;
    float* alpha = ws;
    float* beta  = ws + Bsz * Ksz;
    float* kappa = ws + 2 * Bsz * Ksz;

    ga_zero_w<<<(Bsz * Csz + 255) / 256, 256, 0, stream>>>(w_out);

    ga_params_wmma<<<(Bsz / 16) * 2, 32, 0, stream>>>(
        h_t, W, bias, k_tm1, alpha, beta, kappa, kappa_out);

    ga_phi<<<dim3(Bsz, Tsz / 256), 256, 0, stream>>>(alpha, beta, kappa, phi_out);

    ga_wsum<<<dim3(Bsz, Tsz / 256), 128, 0, stream>>>(phi_out, ctx, w_out);
}